// TrainNhppNm_66907000537733
// MI455X (gfx1250) — compile-verified
//
#include <hip/hip_runtime.h>

typedef __attribute__((ext_vector_type(2))) float v2f;
typedef __attribute__((ext_vector_type(4))) float v4f;
typedef __attribute__((ext_vector_type(8))) float v8f;

#define NN      2048
#define DD      16
#define S_STEPS 10
#define E_EV    4096
#define K_EV    128
#define TSTEP   0.1f
#define NTILE   128   // NN / 16
#define NSLOTS  256

// ---------------- init: zero the f64 accumulator slots ----------------
__global__ void init_kernel(double* acc) {
    int i = blockIdx.x * blockDim.x + threadIdx.x;
    if (i < 2 * NSLOTS) acc[i] = 0.0;
}

// ---------------- x[s,n,:] = z0[0] + t z0[1] + (t^2/2) z0[2]; sq[s,n] = ||x||^2 ----
__global__ void compute_x_kernel(const float* __restrict__ z0,
                                 float* __restrict__ xbuf,
                                 float* __restrict__ sqbuf) {
    int idx = blockIdx.x * blockDim.x + threadIdx.x;   // s*NN + n
    if (idx >= S_STEPS * NN) return;
    int s = idx / NN, n = idx % NN;
    float t  = TSTEP * (float)s;
    float c1 = t;
    float c2 = 0.5f * t * t;
    const float* p0 = z0 + (size_t)n * DD;
    const float* p1 = z0 + ((size_t)NN + n) * DD;
    const float* p2 = z0 + ((size_t)2 * NN + n) * DD;
    float* xo = xbuf + (size_t)idx * DD;
    float acc = 0.f;
#pragma unroll
    for (int d = 0; d < DD; d += 4) {
        v4f a = *(const v4f*)(p0 + d);
        v4f b = *(const v4f*)(p1 + d);
        v4f c = *(const v4f*)(p2 + d);
        v4f x = a + c1 * b + c2 * c;
        *(v4f*)(xo + d) = x;
        acc += x.x * x.x + x.y * x.y + x.z * x.z + x.w * x.w;
    }
    sqbuf[idx] = acc;
}

// ---------------- integral: one wave per 16x16 tile per time step ------------
// Gram tile via V_WMMA_F32_16X16X4_F32 (K accumulated 4 at a time, D=16 -> 4 WMMAs)
__global__ __launch_bounds__(32)
void integral_kernel(const float* __restrict__ gamma,
                     const float* __restrict__ xbuf,
                     const float* __restrict__ sqbuf,
                     double* __restrict__ acc) {
    int rt = blockIdx.x, ct = blockIdx.y, s = blockIdx.z;
    if (ct < rt) return;                  // uniform: EXEC stays all-ones for WMMA
    int lane = threadIdx.x;
    int half = lane >> 4;                 // lane group 0/1 -> K pair {0,1}/{2,3}
    int l16  = lane & 15;

    const float* xs = xbuf + (size_t)s * NN * DD;
    // A: lane%16 = M (row node), half selects K offset; B mirrors with N (col node)
    const float* xr = xs + (size_t)(rt * 16 + l16) * DD + half * 2;
    const float* xc = xs + (size_t)(ct * 16 + l16) * DD + half * 2;

    v8f c = {};
#pragma unroll
    for (int k0 = 0; k0 < DD; k0 += 4) {
        v2f a = *(const v2f*)(xr + k0);
        v2f b = *(const v2f*)(xc + k0);
        c = __builtin_amdgcn_wmma_f32_16x16x4_f32(false, a, false, b,
                                                  (short)0, c, false, false);
    }

    const float* sqs = sqbuf + (size_t)s * NN;
    int   ncol = ct * 16 + l16;           // C/D layout: N = lane%16
    float sqn  = sqs[ncol];
    float gn   = gamma[ncol];
    float part = 0.f;
#pragma unroll
    for (int r = 0; r < 8; ++r) {         // C/D layout: M = r + 8*(lane/16)
        int m = rt * 16 + r + half * 8;
        if (ncol > m) {                   // strict upper triangle
            float d2   = sqs[m] + sqn - 2.0f * c[r];
            float dist = sqrtf(fmaxf(d2, 0.f));
            part += expf(gamma[m] + gn - dist);
        }
    }
    // wave32 butterfly reduction
#pragma unroll
    for (int off = 16; off > 0; off >>= 1) part += __shfl_xor(part, off);
    if (lane == 0) {
        int slot = (rt * 131 + ct * 7 + s) & (NSLOTS - 1);
        atomicAdd(acc + slot, (double)part);
    }
}

// ---------------- event term: one 128-thread block per event -----------------
__global__ __launch_bounds__(128)
void event_kernel(const float* __restrict__ gamma,
                  const float* __restrict__ z0,
                  const float* __restrict__ times,
                  const int* __restrict__ pairs,
                  double* __restrict__ acc) {
    __shared__ float dz[3][DD];
    __shared__ float wpart[4];
    int e   = blockIdx.x;
    int tid = threadIdx.x;
    int i = pairs[2 * e + 0];
    int j = pairs[2 * e + 1];
    if (tid < 48) {
        int o = tid >> 4, d = tid & 15;
        float f = (o == 2) ? 0.5f : 1.0f;   // fact = [1,1,2]
        dz[o][d] = (z0[((size_t)o * NN + i) * DD + d] -
                    z0[((size_t)o * NN + j) * DD + d]) * f;
    }
    __syncthreads();

    float t  = times[(size_t)e * K_EV + tid];
    float t2 = t * t;
    float ss = 0.f;
#pragma unroll
    for (int d = 0; d < DD; ++d) {
        float v = dz[0][d] + t * dz[1][d] + t2 * dz[2][d];
        ss += v * v;
    }
    float ed = sqrtf(ss);
#pragma unroll
    for (int off = 16; off > 0; off >>= 1) ed += __shfl_xor(ed, off);
    if ((tid & 31) == 0) wpart[tid >> 5] = ed;
    __syncthreads();
    if (tid == 0) {
        double tot  = (double)(wpart[0] + wpart[1] + wpart[2] + wpart[3]);
        double gsum = (double)gamma[i] + (double)gamma[j];
        // -log_int[e] = sum_k ed - K * gsum
        atomicAdd(acc + NSLOTS + (e & (NSLOTS - 1)), tot - (double)K_EV * gsum);
    }
}

// ---------------- finalize: out = step * sum(intens) + sum(event terms) ------
__global__ void finalize_kernel(const double* __restrict__ acc, float* __restrict__ out) {
    if (blockIdx.x == 0 && threadIdx.x == 0) {
        double s0 = 0.0, s1 = 0.0;
        for (int i = 0; i < NSLOTS; ++i) s0 += acc[i];
        for (int i = 0; i < NSLOTS; ++i) s1 += acc[NSLOTS + i];
        out[0] = (float)(s0 * (double)TSTEP + s1);
    }
}

extern "C" void kernel_launch(void* const* d_in, const int* in_sizes, int n_in,
                              void* d_out, int out_size, void* d_ws, size_t ws_size,
                              hipStream_t stream) {
    (void)in_sizes; (void)n_in; (void)out_size; (void)ws_size;
    const float* gamma = (const float*)d_in[0];     // (N,)
    const float* z0    = (const float*)d_in[1];     // (O, N, D)
    const float* times = (const float*)d_in[2];     // (E, K)
    const int*   pairs = (const int*)d_in[3];       // (E, 2)
    float* out = (float*)d_out;

    char*   ws    = (char*)d_ws;
    double* acc   = (double*)ws;                                   // 2*NSLOTS doubles (4 KB)
    float*  xbuf  = (float*)(ws + 2 * NSLOTS * sizeof(double));    // S*N*D floats (1.25 MB)
    float*  sqbuf = xbuf + (size_t)S_STEPS * NN * DD;              // S*N floats (80 KB)

    init_kernel<<<(2 * NSLOTS + 255) / 256, 256, 0, stream>>>(acc);
    compute_x_kernel<<<(S_STEPS * NN + 255) / 256, 256, 0, stream>>>(z0, xbuf, sqbuf);

    dim3 grid(NTILE, NTILE, S_STEPS);
    integral_kernel<<<grid, 32, 0, stream>>>(gamma, xbuf, sqbuf, acc);

    event_kernel<<<E_EV, 128, 0, stream>>>(gamma, z0, times, pairs, acc);

    finalize_kernel<<<1, 32, 0, stream>>>(acc, out);
}